// DiffusionGraphConv_34711925686447
// MI455X (gfx1250) — compile-verified
//
#include <hip/hip_runtime.h>

// ---------------------------------------------------------------------------
// DiffusionGraphConv for MI455X (gfx1250, wave32, WMMA).
//   out[b,n,:] = bias + sum_k (D^-1 A)^k-hop features @ W_f[k]  (forward)
//                     + ((D^-1 A)^T)^k-hop features @ W_b[k]    (backward)
// SpMM is edge-parallel with hardware f32 global atomics (L2-resident: the
// 102 MB feature matrix fits in the 192 MB L2). The dense 400000x64 @ 64x64
// transforms run on V_WMMA_F32_16X16X4_F32 (exact fp32 matrix pipe).
// ---------------------------------------------------------------------------

typedef float v2f __attribute__((ext_vector_type(2)));
typedef float v8f __attribute__((ext_vector_type(8)));

#define BB 8          // batch
#define CC 64         // channels in/out
#define FF (BB * CC)  // 512 features per node row

// ---------------- degree / normalization ----------------
__global__ __launch_bounds__(256) void init_deg(float* deg, int n) {
  int i = blockIdx.x * blockDim.x + threadIdx.x;
  if (i < n) deg[i] = 1e-8f;
}

__global__ __launch_bounds__(256) void acc_deg(const int* __restrict__ rows,
                                               const float* __restrict__ ev,
                                               float* __restrict__ deg, int e) {
  int i = blockIdx.x * blockDim.x + threadIdx.x;
  if (i < e) unsafeAtomicAdd(&deg[rows[i]], ev[i]);
}

__global__ __launch_bounds__(256) void calc_nv(const int* __restrict__ rows,
                                               const float* __restrict__ ev,
                                               const float* __restrict__ deg,
                                               float* __restrict__ nv, int e) {
  int i = blockIdx.x * blockDim.x + threadIdx.x;
  if (i < e) nv[i] = ev[i] / deg[rows[i]];
}

// ---------------- buffer init ----------------
__global__ __launch_bounds__(256) void zero_f32(float* p, long long n) {
  long long i = (long long)blockIdx.x * blockDim.x + threadIdx.x;
  if (i < n) p[i] = 0.0f;
}

__global__ __launch_bounds__(256) void init_out_bias(float* __restrict__ out,
                                                     const float* __restrict__ bias,
                                                     long long n) {
  long long i = (long long)blockIdx.x * blockDim.x + threadIdx.x;
  if (i < n) out[i] = bias[i & (CC - 1)];
}

// ---------------- SpMM: one wave per edge, 512 feats ----------------
// dst[scat[e], f] += nv[e] * src_node_major[gath[e], f]
__global__ __launch_bounds__(256) void spmm_from_h(const int* __restrict__ gath,
                                                   const int* __restrict__ scat,
                                                   const float* __restrict__ nv,
                                                   const float* __restrict__ src,
                                                   float* __restrict__ dst, int e) {
  int wave = (blockIdx.x * blockDim.x + threadIdx.x) >> 5;
  int lane = threadIdx.x & 31;
  if (wave >= e) return;
  int g = gath[wave], s = scat[wave];
  float w = nv[wave];
  const float4* srow = (const float4*)(src + (size_t)g * FF);
  float* drow = dst + (size_t)s * FF;
#pragma unroll
  for (int j = 0; j < 4; ++j) {
    int f4 = j * 32 + lane;  // 128 float4s per row
    float4 v = srow[f4];
    int f = f4 * 4;
    unsafeAtomicAdd(drow + f + 0, w * v.x);
    unsafeAtomicAdd(drow + f + 1, w * v.y);
    unsafeAtomicAdd(drow + f + 2, w * v.z);
    unsafeAtomicAdd(drow + f + 3, w * v.w);
  }
}

// Same, but gathers directly from x laid out (B, N, C) -> feature f=b*64+c.
__global__ __launch_bounds__(256) void spmm_from_x(const int* __restrict__ gath,
                                                   const int* __restrict__ scat,
                                                   const float* __restrict__ nv,
                                                   const float* __restrict__ x,
                                                   float* __restrict__ dst,
                                                   int e, int nnodes) {
  int wave = (blockIdx.x * blockDim.x + threadIdx.x) >> 5;
  int lane = threadIdx.x & 31;
  if (wave >= e) return;
  int g = gath[wave], s = scat[wave];
  float w = nv[wave];
  float* drow = dst + (size_t)s * FF;
#pragma unroll
  for (int j = 0; j < 4; ++j) {
    int f4 = j * 32 + lane;
    int f = f4 * 4;           // aligned-4 group never crosses a 64-channel block
    int b = f >> 6, c = f & 63;
    float4 v = *(const float4*)(x + (size_t)b * nnodes * CC + (size_t)g * CC + c);
    unsafeAtomicAdd(drow + f + 0, w * v.x);
    unsafeAtomicAdd(drow + f + 1, w * v.y);
    unsafeAtomicAdd(drow + f + 2, w * v.z);
    unsafeAtomicAdd(drow + f + 3, w * v.w);
  }
}

// ---------------- dense transform on the WMMA pipe ----------------
// out(b,n,:) += h_row(n*B+b, :) @ W ; one wave computes a 16x16 tile of the
// (Nnodes*B x 64) @ (64 x 64) product with 16 chained V_WMMA_F32_16X16X4_F32.
__global__ __launch_bounds__(256) void gemm_acc_wmma(const float* __restrict__ h,
                                                     const float* __restrict__ W,
                                                     float* __restrict__ out,
                                                     int nnodes, int mtiles) {
  int wave = (blockIdx.x * blockDim.x + threadIdx.x) >> 5;
  int lane = threadIdx.x & 31;
  int ntile = wave & 3;   // 64 cols -> 4 tiles
  int mtile = wave >> 2;
  if (mtile >= mtiles) return;  // wave-uniform: EXEC stays all-1s for WMMA
  int r0 = mtile << 4;
  int n0 = ntile << 4;
  int hi = lane >> 4;     // half-wave selector
  int lo = lane & 15;

  // C: VGPR v holds M=v (lanes 0-15) / M=v+8 (lanes 16-31), N=lane%16.
  v8f c;
  int mbase = r0 + hi * 8;
#pragma unroll
  for (int v = 0; v < 8; ++v) {
    int r = mbase + v;              // GEMM row = node*B + b
    int nn = r >> 3, bb = r & 7;
    c[v] = out[(size_t)bb * nnodes * CC + (size_t)nn * CC + n0 + lo];
  }

#pragma unroll
  for (int kk = 0; kk < CC; kk += 4) {
    int ka = kk + hi * 2;
    // A (16x4): lanes 0-15 -> {K=kk,kk+1}, lanes 16-31 -> {K=kk+2,kk+3}
    v2f a, b;
    const float* arow = h + (size_t)(r0 + lo) * CC + ka;
    a.x = arow[0];
    a.y = arow[1];
    // B (4x16): mirrored half-wave K striping, N = lane%16
    b.x = W[ka * CC + n0 + lo];
    b.y = W[(ka + 1) * CC + n0 + lo];
    c = __builtin_amdgcn_wmma_f32_16x16x4_f32(false, a, false, b,
                                              (short)0, c, false, false);
  }

#pragma unroll
  for (int v = 0; v < 8; ++v) {
    int r = mbase + v;
    int nn = r >> 3, bb = r & 7;
    out[(size_t)bb * nnodes * CC + (size_t)nn * CC + n0 + lo] = c[v];
  }
}

// ---------------------------------------------------------------------------
extern "C" void kernel_launch(void* const* d_in, const int* in_sizes, int n_in,
                              void* d_out, int out_size, void* d_ws, size_t ws_size,
                              hipStream_t stream) {
  const float* x    = (const float*)d_in[0];
  const int*   ei   = (const int*)d_in[1];
  const float* ev   = (const float*)d_in[2];
  const float* Wf   = (const float*)d_in[3];  // (2,64,64)
  const float* Wb   = (const float*)d_in[4];  // (2,64,64)
  const float* bias = (const float*)d_in[5];
  float* out = (float*)d_out;

  const int N = in_sizes[0] / FF;       // nodes
  const int E = in_sizes[2];            // edges
  const int* rows = ei;
  const int* cols = ei + E;

  // workspace: deg[N] | nv[E] | hA[N*512] | hB[N*512]
  float* ws  = (float*)d_ws;
  float* deg = ws;
  float* nv  = deg + N;
  float* hA  = nv + E;
  float* hB  = hA + (size_t)N * FF;

  const long long outN = (long long)BB * N * CC;
  const long long hN   = (long long)N * FF;
  const int mtiles = (N * BB) / 16;               // 25000
  const int gemmBlocks = (mtiles * 4 + 7) / 8;    // 8 waves/block, 1 tile/wave
  const int spmmBlocks = (E + 7) / 8;             // 8 edges/block

  // normalization
  init_deg<<<(N + 255) / 256, 256, 0, stream>>>(deg, N);
  acc_deg<<<(E + 255) / 256, 256, 0, stream>>>(rows, ev, deg, E);
  calc_nv<<<(E + 255) / 256, 256, 0, stream>>>(rows, ev, deg, nv, E);

  // out = bias
  init_out_bias<<<(int)((outN + 255) / 256), 256, 0, stream>>>(out, bias, outN);

  // ---- forward diffusion: gather cols, scatter rows ----
  zero_f32<<<(int)((hN + 255) / 256), 256, 0, stream>>>(hA, hN);
  spmm_from_x<<<spmmBlocks, 256, 0, stream>>>(cols, rows, nv, x, hA, E, N);
  gemm_acc_wmma<<<gemmBlocks, 256, 0, stream>>>(hA, Wf + 0 * CC * CC, out, N, mtiles);

  zero_f32<<<(int)((hN + 255) / 256), 256, 0, stream>>>(hB, hN);
  spmm_from_h<<<spmmBlocks, 256, 0, stream>>>(cols, rows, nv, hA, hB, E);
  gemm_acc_wmma<<<gemmBlocks, 256, 0, stream>>>(hB, Wf + 1 * CC * CC, out, N, mtiles);

  // ---- backward diffusion: gather rows, scatter cols ----
  zero_f32<<<(int)((hN + 255) / 256), 256, 0, stream>>>(hA, hN);
  spmm_from_x<<<spmmBlocks, 256, 0, stream>>>(rows, cols, nv, x, hA, E, N);
  gemm_acc_wmma<<<gemmBlocks, 256, 0, stream>>>(hA, Wb + 0 * CC * CC, out, N, mtiles);

  zero_f32<<<(int)((hN + 255) / 256), 256, 0, stream>>>(hB, hN);
  spmm_from_h<<<spmmBlocks, 256, 0, stream>>>(rows, cols, nv, hA, hB, E);
  gemm_acc_wmma<<<gemmBlocks, 256, 0, stream>>>(hB, Wb + 1 * CC * CC, out, N, mtiles);
}